// RandomMotionBlur_31207232373083
// MI455X (gfx1250) — compile-verified
//
#include <hip/hip_runtime.h>
#include <hip/hip_bf16.h>

#define KS   17
#define PADH 8
#define HGT  4096
#define WID  4096
#define ALPHA_C 0.2f
#define EPS_C   1e-8f

#define PITCH   84            // LDS patch pitch (floats): bank-stride 20 mod 64 -> conflict-free
#define PATROWS 80            // 64 output rows + 2*8 halo
#define PATC4   20            // 80 floats per row / 4
#define NBW     (KS * 8 * 32 * 2)   // banded-B operand pool: 8704 floats

typedef __attribute__((ext_vector_type(2))) float v2f;
typedef __attribute__((ext_vector_type(4))) float v4f;
typedef __attribute__((ext_vector_type(8))) float v8f;
typedef __attribute__((ext_vector_type(4))) int   v4i;

#define AS1 __attribute__((address_space(1)))
#define AS3 __attribute__((address_space(3)))

#if __has_builtin(__builtin_amdgcn_global_load_async_to_lds_b128) && \
    __has_builtin(__builtin_amdgcn_s_wait_asynccnt)
#define HAVE_ASYNC_LDS 1
#else
#define HAVE_ASYNC_LDS 0
#endif

#if HAVE_ASYNC_LDS
// Generic LDS pointers carry the LDS byte offset in their low 32 bits
// (aperture lives in the high half), so integer truncation yields the AS3 ptr.
__device__ __forceinline__ AS3 v4i* to_lds(const void* p) {
  return (AS3 v4i*)(unsigned int)(unsigned long long)p;
}
__device__ __forceinline__ AS1 v4i* to_glb(const void* p) {
  return (AS1 v4i*)(unsigned long long)p;
}
#endif

// ---------------------------------------------------------------------------
// Phase 1: synthesize the 17x17 motion kernel (reference scan + bilinear
// scatter + normalize), then expand it into 17 banded 32x16 WMMA-B operand
// matrices B_dy[k][n] = w[dy][k-n], pre-packed in the exact per-lane VGPR
// layout of V_WMMA_F32_16X16X4_F32's B operand.
// ---------------------------------------------------------------------------
__global__ void mb_build_kernel(const float* __restrict__ init_re,
                                const float* __restrict__ init_im,
                                const float* __restrict__ change_re,
                                const float* __restrict__ change_im,
                                float* __restrict__ Bws) {
  __shared__ float sw[KS * KS];
  const int tid = threadIdx.x;

  if (tid == 0) {
    float vre = init_re[0], vim = init_im[0];
    float pre = 0.f, pim = 0.f;
    float mre[KS + 1], mim[KS + 1];
    mre[0] = 0.f; mim[0] = 0.f;
    for (int s = 0; s < KS; ++s) {          // change_* shape [KS, B=1]
      vre += change_re[s] * ALPHA_C;
      vim += change_im[s] * ALPHA_C;
      float mag = sqrtf(vre * vre + vim * vim) + EPS_C;
      vre /= mag; vim /= mag;
      pre += vre; pim += vim;
      mre[s + 1] = pre; mim[s + 1] = pim;
    }
    float rmin = mre[0], rmax = mre[0], imin = mim[0], imax = mim[0];
    for (int i = 1; i <= KS; ++i) {
      rmin = fminf(rmin, mre[i]); rmax = fmaxf(rmax, mre[i]);
      imin = fminf(imin, mim[i]); imax = fmaxf(imax, mim[i]);
    }
    float scale = fminf((float)(KS - 1) / (rmax - rmin),
                        (float)(KS - 1) / (imax - imin));
    float rshift = ((float)KS - (rmax - rmin) * scale) * 0.5f - rmin * scale;
    float ishift = ((float)KS - (imax - imin) * scale) * 0.5f - imin * scale;

    float w[KS * KS];
    for (int i = 0; i < KS * KS; ++i) w[i] = 0.f;
    const float inv = 1.0f / (float)KS;
    for (int i = 0; i <= KS; ++i) {
      float xs = fminf(fmaxf(mre[i] * scale + rshift, 0.f), (float)(KS - 1));
      float ys = fminf(fmaxf(mim[i] * scale + ishift, 0.f), (float)(KS - 1));
      int ix = (int)xs, iy = (int)ys;
      float vx = xs - (float)ix, vy = ys - (float)iy;
      int ix1 = min(ix + 1, KS - 1), iy1 = min(iy + 1, KS - 1);
      bool mx = (ix + 1) < KS, my = (iy + 1) < KS;
      w[iy  * KS + ix ] += (1.f - vx) * (1.f - vy) * inv;
      w[iy  * KS + ix1] += mx ? vx * (1.f - vy) * inv : 0.f;
      w[iy1 * KS + ix ] += my ? (1.f - vx) * vy * inv : 0.f;
      w[iy1 * KS + ix1] += (mx && my) ? vx * vy * inv : 0.f;
    }
    float s = 0.f;
    for (int i = 0; i < KS * KS; ++i) s += w[i];
    float rs = 1.f / s;
    for (int i = 0; i < KS * KS; ++i) sw[i] = w[i] * rs;
  }
  __syncthreads();

  // Pack: index i = dy*512 + kq*64 + L*2 + g
  // lane L holds B[k][n] with n = L&15, k = kq*4 + 2*(L>>4) + g
  for (int i = tid; i < NBW; i += blockDim.x) {
    int g  = i & 1;
    int L  = (i >> 1) & 31;
    int kq = (i >> 6) & 7;
    int dy = i >> 9;
    int k = kq * 4 + 2 * (L >> 4) + g;
    int n = L & 15;
    int d = k - n;
    Bws[i] = ((unsigned)d < (unsigned)KS) ? sw[dy * KS + d] : 0.f;
  }
}

// ---------------------------------------------------------------------------
// Phase 2: depthwise 17x17 conv as banded GEMMs on V_WMMA_F32_16X16X4_F32.
// Workgroup = 128 threads (4 waves) covering a 64x64 output block of one
// channel. LDS: banded-B pool (34816 B) + 80x80 input patch w/ halo
// (pitch 84 -> 26880 B) = 61696 B.
// Interior blocks stage via GLOBAL_LOAD_ASYNC_TO_LDS (ASYNCcnt); border
// blocks fall back to masked VGPR staging with zero padding.
// Each wave: 16-row x 64-col strip -> 4 accumulators, 17*8*4 = 544 WMMAs.
// ---------------------------------------------------------------------------
__global__ __launch_bounds__(128)
void mb_conv_wmma(const float* __restrict__ x,
                  const float* __restrict__ Bws,
                  float* __restrict__ out) {
  __shared__ __align__(16) float sB[NBW];
  __shared__ __align__(16) float sP[PATROWS * PITCH];

  const int tid  = threadIdx.x;
  const int lane = tid & 31;
  const int wv   = tid >> 5;
  const int lm   = lane & 15;
  const int hi   = lane >> 4;

  const int x0   = blockIdx.x * 64;
  const int y0wg = blockIdx.y * 64;
  const int c    = blockIdx.z;
  const float* __restrict__ xc = x + (size_t)c * HGT * WID;

  // Patch fully inside the image? (uniform per block)
  const bool interior = (x0 >= PADH) && (x0 + 71 < WID) &&
                        (y0wg >= PADH) && (y0wg + 71 < HGT);

#if HAVE_ASYNC_LDS
  if (interior) {
    // banded weights: 2176 x b128, global -> LDS without touching VGPRs
    for (int i = tid; i < NBW / 4; i += 128)
      __builtin_amdgcn_global_load_async_to_lds_b128(
          to_glb((const v4f*)Bws + i), to_lds((const v4f*)sB + i), 0, 0);
    // 80x80 input patch
    for (int i = tid; i < PATROWS * PATC4; i += 128) {
      int r  = i / PATC4;
      int c4 = i % PATC4;
      int gy = y0wg - PADH + r;
      int gx = x0   - PADH + c4 * 4;
      __builtin_amdgcn_global_load_async_to_lds_b128(
          to_glb(xc + (size_t)gy * WID + gx),
          to_lds(sP + r * PITCH + c4 * 4), 0, 0);
    }
    __builtin_amdgcn_s_wait_asynccnt(0);
  } else
#endif
  {
    // --- stage banded weights (coalesced b128 via VGPRs) ---
    for (int i = tid; i < NBW / 4; i += 128)
      ((v4f*)sB)[i] = ((const v4f*)Bws)[i];

    // --- stage 80x80 input patch with halo, zero-padded at image borders ---
    for (int i = tid; i < PATROWS * PATC4; i += 128) {
      int r  = i / PATC4;
      int c4 = i % PATC4;
      int gy = y0wg - PADH + r;
      int gx = x0   - PADH + c4 * 4;   // x0 % 64 == 0 -> gx % 4 == 0 (aligned)
      v4f v;
      if (gy >= 0 && gy < HGT && gx >= 0 && gx + 3 < WID) {
        v = *(const v4f*)&xc[(size_t)gy * WID + gx];
      } else {
        #pragma unroll
        for (int j = 0; j < 4; ++j) {
          int gxx = gx + j;
          v[j] = (gy >= 0 && gy < HGT && gxx >= 0 && gxx < WID)
                   ? xc[(size_t)gy * WID + gxx] : 0.f;
        }
      }
      *(v4f*)&sP[r * PITCH + c4 * 4] = v;
    }
  }
  __syncthreads();

  // Warm L2 for the block one strip below while we compute (hint only).
  {
    int py = y0wg + 64 + (tid >> 1);             // 64 rows, 2 threads/row
    int px = max(x0 - PADH, 0) + (tid & 1) * 48; // ~2 cachelines per row
    if (py < HGT)
      __builtin_prefetch(&xc[(size_t)py * WID + px], 0, 1);
  }

  const int y0     = y0wg + wv * 16;
  const int rowoff = wv * 16;

  v8f D0 = {}, D1 = {}, D2 = {}, D3 = {};

  for (int dy = 0; dy < KS; ++dy) {
    // B fragments for this dy (8 chunks of 4x16, reused by all 4 tiles)
    v2f b[8];
    #pragma unroll
    for (int kq = 0; kq < 8; ++kq)
      b[kq] = *(const v2f*)&sB[((dy * 8 + kq) * 32 + lane) * 2];

    // A fragments: A[m][k] = patch[rowoff + dy + m][16*t + kq*4 + k]
    // lane layout: M = lane&15, K pair = {base, base+1}, base = kq*4 + 2*(lane>>4)
    const float* prow = &sP[(rowoff + dy + lm) * PITCH + 2 * hi];

    #pragma unroll
    for (int kq = 0; kq < 8; ++kq) {
      const float* p = prow + kq * 4;
      v2f a0 = *(const v2f*)(p +  0);
      v2f a1 = *(const v2f*)(p + 16);
      v2f a2 = *(const v2f*)(p + 32);
      v2f a3 = *(const v2f*)(p + 48);
      D0 = __builtin_amdgcn_wmma_f32_16x16x4_f32(false, a0, false, b[kq],
                                                 (short)0, D0, false, false);
      D1 = __builtin_amdgcn_wmma_f32_16x16x4_f32(false, a1, false, b[kq],
                                                 (short)0, D1, false, false);
      D2 = __builtin_amdgcn_wmma_f32_16x16x4_f32(false, a2, false, b[kq],
                                                 (short)0, D2, false, false);
      D3 = __builtin_amdgcn_wmma_f32_16x16x4_f32(false, a3, false, b[kq],
                                                 (short)0, D3, false, false);
    }
  }

  // --- store: D VGPR r holds rows y0+r (lanes 0-15) and y0+r+8 (lanes 16-31)
  float* __restrict__ oc = out + (size_t)c * HGT * WID;
  #pragma unroll
  for (int r = 0; r < 8; ++r) {
    int gy = y0 + r + 8 * hi;
    size_t base = (size_t)gy * WID + (size_t)(x0 + lm);
    oc[base +  0] = D0[r];
    oc[base + 16] = D1[r];
    oc[base + 32] = D2[r];
    oc[base + 48] = D3[r];
  }
}

extern "C" void kernel_launch(void* const* d_in, const int* in_sizes, int n_in,
                              void* d_out, int out_size, void* d_ws, size_t ws_size,
                              hipStream_t stream) {
  const float* x         = (const float*)d_in[0];
  const float* init_re   = (const float*)d_in[1];
  const float* init_im   = (const float*)d_in[2];
  const float* change_re = (const float*)d_in[3];
  const float* change_im = (const float*)d_in[4];
  float* Bws = (float*)d_ws;   // 8704 floats (34816 bytes) of banded-B operands

  mb_build_kernel<<<1, 256, 0, stream>>>(init_re, init_im, change_re, change_im, Bws);

  dim3 grid(WID / 64, HGT / 64, 3);
  mb_conv_wmma<<<grid, 128, 0, stream>>>(x, Bws, (float*)d_out);
}